// GAT_loop_42640435315482
// MI455X (gfx1250) — compile-verified
//
#include <hip/hip_runtime.h>
#include <hip/hip_bf16.h>
#include <math.h>

// ---------------- problem constants (from reference) ----------------
#define NN    50000      // nodes
#define EE    800000     // edges (before self loops)
#define ET    (EE + NN)  // edges with self loops
#define FH    128        // F_IN == H*C == 128
#define HH    4          // heads
#define CC    32         // per-head channels
#define GG    256        // graphs

typedef float v2f  __attribute__((ext_vector_type(2)));
typedef float v8f  __attribute__((ext_vector_type(8)));

// ---------------- helpers ----------------
__device__ __forceinline__ void atomicMaxF(float* addr, float val) {
    // monotonic bit trick: works for mixed-sign floats
    if (val >= 0.0f) atomicMax((int*)addr, __float_as_int(val));
    else             atomicMin((unsigned int*)addr, __float_as_uint(val));
}

__global__ void fill_f32(float* __restrict__ p, float v, int n) {
    int i = blockIdx.x * blockDim.x + threadIdx.x;
    if (i < n) p[i] = v;
}

// ---------------- WMMA fp32 GEMM: H = X[N,128] * W[128,128] ----------------
// One block = 256 threads = 8 wave32 waves. Block owns a 16-row strip of X
// (staged in LDS, padded stride 132 to avoid bank conflicts on column reads).
// Wave w computes the 16x16 output tile for columns [16w,16w+16) using
// V_WMMA_F32_16X16X4_F32, K unrolled in 32 steps of 4.
__global__ void __launch_bounds__(256)
gemm_xw_wmma(const float* __restrict__ X, const float* __restrict__ W,
             float* __restrict__ Hout) {
    __shared__ float As[16][132];
    const int r0 = blockIdx.x * 16;   // N = 50000 = 16*3125, exact

    for (int i = threadIdx.x; i < 16 * FH; i += 256) {
        int r = i >> 7, c = i & 127;
        As[r][c] = X[(size_t)(r0 + r) * FH + c];
    }
    __syncthreads();

    const int wave = threadIdx.x >> 5;
    const int lane = threadIdx.x & 31;
    const int j0   = wave * 16;
    const int m    = lane & 15;          // A row / D column-group lane id
    const int kh   = (lane >> 4) * 2;    // K sub-offset: lanes16-31 hold K+2
    v8f acc = {};

#pragma unroll
    for (int kk = 0; kk < FH; kk += 4) {
        v2f a, b;
        // A 16x4 fp32 fragment: VGPR v holds K = kk + kh + v for row m
        a.x = As[m][kk + kh + 0];
        a.y = As[m][kk + kh + 1];
        // B 4x16 fp32 fragment (symmetric layout): row K across lanes (col j0+m)
        b.x = W[(size_t)(kk + kh + 0) * FH + j0 + m];
        b.y = W[(size_t)(kk + kh + 1) * FH + j0 + m];
        acc = __builtin_amdgcn_wmma_f32_16x16x4_f32(
            /*neg_a=*/false, a, /*neg_b=*/false, b,
            /*c_mod=*/(short)0, acc, /*reuse_a=*/false, /*reuse_b=*/false);
    }

    // D 16x16 f32: lanes 0-15 -> rows v (v=0..7), lanes 16-31 -> rows v+8
    const int rbase = (lane >> 4) * 8;
#pragma unroll
    for (int v = 0; v < 8; ++v)
        Hout[(size_t)(r0 + rbase + v) * FH + j0 + m] = acc[v];
}

// ---------------- attention logits per node/head ----------------
__global__ void alpha_kernel(const float* __restrict__ h,
                             const float* __restrict__ a_src,
                             const float* __restrict__ a_dst,
                             float* __restrict__ als, float* __restrict__ ald) {
    int idx = blockIdx.x * blockDim.x + threadIdx.x;
    if (idx >= NN * HH) return;
    int n = idx >> 2, hd = idx & 3;
    const float* hp = h + (size_t)n * FH + hd * CC;
    const float* s  = a_src + hd * CC;
    const float* d  = a_dst + hd * CC;
    float vs = 0.f, vd = 0.f;
#pragma unroll
    for (int c = 0; c < CC; ++c) { float x = hp[c]; vs += x * s[c]; vd += x * d[c]; }
    als[idx] = vs;
    ald[idx] = vd;
}

// ---------------- edge pass 1: leaky-relu logits + segment max ----------------
__global__ void edge_logits_max(const int* __restrict__ ei,
                                const float* __restrict__ als,
                                const float* __restrict__ ald,
                                float* __restrict__ ebuf,
                                float* __restrict__ emax) {
    int e = blockIdx.x * blockDim.x + threadIdx.x;
    if (e >= ET) return;
    int s, d;
    if (e < EE) { s = ei[e]; d = ei[EE + e]; } else { s = d = e - EE; }
    const float4 av = *(const float4*)(als + (size_t)s * 4);
    const float4 dv = *(const float4*)(ald + (size_t)d * 4);
    float v[4] = {av.x + dv.x, av.y + dv.y, av.z + dv.z, av.w + dv.w};
#pragma unroll
    for (int hd = 0; hd < 4; ++hd) {
        float x = v[hd];
        x = x > 0.f ? x : 0.2f * x;          // leaky relu, slope 0.2
        ebuf[(size_t)e * 4 + hd] = x;
        atomicMaxF(&emax[(size_t)d * 4 + hd], x);
    }
}

// ---------------- edge pass 2: exp + denominator ----------------
__global__ void edge_exp_sum(const int* __restrict__ ei,
                             float* __restrict__ ebuf,
                             const float* __restrict__ emax,
                             float* __restrict__ denom) {
    int e = blockIdx.x * blockDim.x + threadIdx.x;
    if (e >= ET) return;
    int d = (e < EE) ? ei[EE + e] : (e - EE);
    const float4 em = *(const float4*)(emax + (size_t)d * 4);
    float m[4] = {em.x, em.y, em.z, em.w};
#pragma unroll
    for (int hd = 0; hd < 4; ++hd) {
        float p = __expf(ebuf[(size_t)e * 4 + hd] - m[hd]);
        ebuf[(size_t)e * 4 + hd] = p;
        atomicAdd(&denom[(size_t)d * 4 + hd], p);
    }
}

// ---------------- edge pass 3: weighted scatter-add of messages ----------------
__global__ void edge_aggregate(const int* __restrict__ ei,
                               const float* __restrict__ ebuf,
                               const float* __restrict__ denom,
                               const float* __restrict__ h,
                               float* __restrict__ outh) {
    int idx = blockIdx.x * blockDim.x + threadIdx.x;  // (edge, head)
    if (idx >= ET * HH) return;
    int e = idx >> 2, hd = idx & 3;
    int s, d;
    if (e < EE) { s = ei[e]; d = ei[EE + e]; } else { s = d = e - EE; }
    float alpha = ebuf[(size_t)e * 4 + hd] / (denom[(size_t)d * 4 + hd] + 1e-16f);
    const float* hs = h    + (size_t)s * FH + hd * CC;
    float*       od = outh + (size_t)d * FH + hd * CC;
#pragma unroll
    for (int c = 0; c < CC; c += 4) {
        float4 hv = *(const float4*)(hs + c);
        atomicAdd(od + c + 0, alpha * hv.x);
        atomicAdd(od + c + 1, alpha * hv.y);
        atomicAdd(od + c + 2, alpha * hv.z);
        atomicAdd(od + c + 3, alpha * hv.w);
    }
}

// ---------------- bias + tanh (in place) ----------------
__global__ void bias_tanh(float* __restrict__ h, const float* __restrict__ b) {
    int i = blockIdx.x * blockDim.x + threadIdx.x;
    if (i >= NN * FH) return;
    h[i] = tanhf(h[i] + b[i & 127]);
}

// ---------------- pooling ----------------
__global__ void pool_scatter(const float* __restrict__ hid,
                             const int* __restrict__ batch,
                             float* __restrict__ gmax, float* __restrict__ gsum) {
    int i = blockIdx.x * blockDim.x + threadIdx.x;
    if (i >= NN * FH) return;
    int n = i >> 7, c = i & 127;
    int g = batch[n];
    float v = hid[i];
    atomicMaxF(&gmax[(size_t)g * FH + c], v);
    atomicAdd(&gsum[(size_t)g * FH + c], v);
}

__global__ void count_nodes(const int* __restrict__ batch, float* __restrict__ cnt) {
    int n = blockIdx.x * blockDim.x + threadIdx.x;
    if (n < NN) atomicAdd(&cnt[batch[n]], 1.0f);
}

// one block per graph: emit pooled row + out = pooled . Wout + bout
__global__ void __launch_bounds__(256)
final_out(const float* __restrict__ gmax, const float* __restrict__ gsum,
          const float* __restrict__ cnt, const float* __restrict__ Wout,
          const float* __restrict__ bout, float* __restrict__ out) {
    __shared__ float red[256];
    int g = blockIdx.x, t = threadIdx.x;
    float cn = fmaxf(cnt[g], 1.0f);
    float p = (t < FH) ? gmax[(size_t)g * FH + t]
                       : gsum[(size_t)g * FH + (t - FH)] / cn;
    out[GG + (size_t)g * 2 * FH + t] = p;   // pooled section (after out[G])
    red[t] = p * Wout[t];
    __syncthreads();
    for (int s = 128; s > 0; s >>= 1) {
        if (t < s) red[t] += red[t + s];
        __syncthreads();
    }
    if (t == 0) out[g] = red[0] + bout[0];
}

// ---------------- host-side orchestration ----------------
static void run_gat_layer(const float* x_in, const float* W, const float* as,
                          const float* ad, const float* b,
                          float* h_tmp, float* out_hidden,
                          float* als, float* ald, float* emax, float* denom,
                          float* ebuf, const int* ei, hipStream_t stream) {
    gemm_xw_wmma<<<NN / 16, 256, 0, stream>>>(x_in, W, h_tmp);
    alpha_kernel<<<(NN * HH + 255) / 256, 256, 0, stream>>>(h_tmp, as, ad, als, ald);
    fill_f32<<<(NN * HH + 255) / 256, 256, 0, stream>>>(emax, -1e30f, NN * HH);
    fill_f32<<<(NN * HH + 255) / 256, 256, 0, stream>>>(denom, 0.0f, NN * HH);
    fill_f32<<<(NN * FH + 255) / 256, 256, 0, stream>>>(out_hidden, 0.0f, NN * FH);
    edge_logits_max<<<(ET + 255) / 256, 256, 0, stream>>>(ei, als, ald, ebuf, emax);
    edge_exp_sum<<<(ET + 255) / 256, 256, 0, stream>>>(ei, ebuf, emax, denom);
    edge_aggregate<<<((size_t)ET * HH + 255) / 256, 256, 0, stream>>>(ei, ebuf, denom, h_tmp, out_hidden);
    bias_tanh<<<(NN * FH + 255) / 256, 256, 0, stream>>>(out_hidden, b);
}

extern "C" void kernel_launch(void* const* d_in, const int* in_sizes, int n_in,
                              void* d_out, int out_size, void* d_ws, size_t ws_size,
                              hipStream_t stream) {
    const float* x     = (const float*)d_in[0];
    const int*   ei    = (const int*)  d_in[1];   // [2,E] flat: row0=src, row1=dst
    const int*   batch = (const int*)  d_in[2];
    const float* W0 = (const float*)d_in[3],  *as0 = (const float*)d_in[4],
               *ad0 = (const float*)d_in[5],  *b0  = (const float*)d_in[6];
    const float* W1 = (const float*)d_in[7],  *as1 = (const float*)d_in[8],
               *ad1 = (const float*)d_in[9],  *b1  = (const float*)d_in[10];
    const float* W2 = (const float*)d_in[11], *as2 = (const float*)d_in[12],
               *ad2 = (const float*)d_in[13], *b2  = (const float*)d_in[14];
    const float* Wout = (const float*)d_in[15], *bout = (const float*)d_in[16];
    float* out = (float*)d_out;

    // workspace carve-up (floats, 256B aligned chunks)
    size_t off = 0;
    auto alloc = [&](size_t nfloats) -> float* {
        float* p = (float*)d_ws + off;
        off += (nfloats + 63) & ~(size_t)63;
        return p;
    };
    float* h_tmp = alloc((size_t)NN * FH);
    float* hid0  = alloc((size_t)NN * FH);
    float* hid1  = alloc((size_t)NN * FH);
    float* als   = alloc((size_t)NN * HH);
    float* ald   = alloc((size_t)NN * HH);
    float* emax  = alloc((size_t)NN * HH);
    float* denom = alloc((size_t)NN * HH);
    float* ebuf  = alloc((size_t)ET * HH);
    float* gmax  = alloc((size_t)GG * FH);
    float* gsum  = alloc((size_t)GG * FH);
    float* cnt   = alloc((size_t)GG);
    (void)ws_size; (void)n_in; (void)in_sizes; (void)out_size;

    // three GAT layers (all dims 128 -> 128)
    run_gat_layer(x,    W0, as0, ad0, b0, h_tmp, hid0, als, ald, emax, denom, ebuf, ei, stream);
    run_gat_layer(hid0, W1, as1, ad1, b1, h_tmp, hid1, als, ald, emax, denom, ebuf, ei, stream);
    run_gat_layer(hid1, W2, as2, ad2, b2, h_tmp, hid0, als, ald, emax, denom, ebuf, ei, stream);

    // global max / mean pooling + output head
    fill_f32<<<(GG * FH + 255) / 256, 256, 0, stream>>>(gmax, -1e30f, GG * FH);
    fill_f32<<<(GG * FH + 255) / 256, 256, 0, stream>>>(gsum, 0.0f, GG * FH);
    fill_f32<<<(GG + 255) / 256, 256, 0, stream>>>(cnt, 0.0f, GG);
    pool_scatter<<<(NN * FH + 255) / 256, 256, 0, stream>>>(hid0, batch, gmax, gsum);
    count_nodes<<<(NN + 255) / 256, 256, 0, stream>>>(batch, cnt);
    final_out<<<GG, 256, 0, stream>>>(gmax, gsum, cnt, Wout, bout, out);
}